// MultiHeadGATLayer_45243185496498
// MI455X (gfx1250) — compile-verified
//
#include <hip/hip_runtime.h>
#include <hip/hip_bf16.h>
#include <math.h>

// ---------------------------------------------------------------------------
// MultiHeadGAT layer for gfx1250.
//   IN=128, H=8, C=32, HC=256.  N = 50000 (multiple of 16), E = 1.6M.
//   GEMM h = x@W via V_WMMA_F32_16X16X4_F32, 16x64 tile per wave (A reuse x4).
// ---------------------------------------------------------------------------

#define HC   256
#define IND  128
#define NHEAD 8
#define CDIM 32

typedef __attribute__((ext_vector_type(2))) float v2f;
typedef __attribute__((ext_vector_type(8))) float v8f;

// ------------------------- helpers -----------------------------------------

__device__ __forceinline__ void atomicMaxF(float* addr, float v) {
  // IEEE-754 monotone bit trick: signed max for v>=0, unsigned min for v<0.
  // -inf (0xFF800000) init works for both paths; lowers to native
  // GLOBAL_ATOMIC_MAX_I32 / GLOBAL_ATOMIC_MIN_U32 (no CAS loop).
  if (v >= 0.0f) atomicMax((int*)addr, __float_as_int(v));
  else           atomicMin((unsigned int*)addr, __float_as_uint(v));
}

// ------------------------- init --------------------------------------------

__global__ void gat_init_kernel(float* __restrict__ amax, float* __restrict__ denom,
                                float* __restrict__ colsum, float* __restrict__ colsq,
                                float* __restrict__ ea_sum, float* __restrict__ edge_coef,
                                const float* __restrict__ W_edge,
                                const float* __restrict__ att_edge, int NH) {
  int t = blockIdx.x * blockDim.x + threadIdx.x;
  for (int i = t; i < NH; i += gridDim.x * blockDim.x) {
    amax[i]  = -INFINITY;
    denom[i] = 0.0f;
  }
  if (blockIdx.x == 0) {
    int tid = threadIdx.x;                  // blockDim.x == 256 required
    if (tid < HC) { colsum[tid] = 0.0f; colsq[tid] = 0.0f; }
    if (tid == 0) ea_sum[0] = 0.0f;
    // edge_coef[h] = sum_c W_edge[0, h*C+c] * att_edge[h, c]
    float p = W_edge[tid] * att_edge[tid];  // tid = h*32 + c; one head per wave32
    for (int off = 16; off > 0; off >>= 1) p += __shfl_down(p, off, 32);
    if ((tid & 31) == 0) edge_coef[tid >> 5] = p;
  }
}

// ------------------------- mean(edge_attr) ---------------------------------

__global__ void gat_ea_reduce_kernel(const float* __restrict__ edge_attr,
                                     float* __restrict__ ea_sum, int E) {
  __shared__ float red[8];
  float s = 0.0f;
  for (int i = blockIdx.x * blockDim.x + threadIdx.x; i < E;
       i += gridDim.x * blockDim.x) s += edge_attr[i];
  for (int off = 16; off > 0; off >>= 1) s += __shfl_down(s, off, 32);
  if ((threadIdx.x & 31) == 0) red[threadIdx.x >> 5] = s;
  __syncthreads();
  if (threadIdx.x == 0) {
    float v = 0.0f;
    #pragma unroll
    for (int w = 0; w < 8; ++w) v += red[w];
    atomicAdd(ea_sum, v);
  }
}

// ------------------------- WMMA GEMM: h = x @ W ----------------------------
// grid = N/16 blocks, block = 128 (4 waves). Wave w computes the 16x64 strip
// [mTile*16 .. +16) x [w*64 .. +64): 4 accumulators share one A fragment per
// k-step (4x less A traffic, 4 independent WMMA dep-chains).
// A frag (16x4 f32, 2 VGPR): lanes 0-15 hold K=k,k+1 for M=lane;
//                            lanes 16-31 hold K=k+2,k+3.
// B frag (4x16 f32, 2 VGPR): lanes 0-15 hold K=k,k+1 for N=lane;
//                            lanes 16-31 hold K=k+2,k+3.
// D (16x16 f32, 8 VGPR):     acc[i] = D[M = i + 8*(lane>=16)][N = lane&15].

__global__ void gat_gemm_wmma_kernel(const float* __restrict__ x,
                                     const float* __restrict__ W,
                                     float* __restrict__ h, int N) {
  const int wave  = threadIdx.x >> 5;
  const int lane  = threadIdx.x & 31;
  const int mTile = blockIdx.x;
  const int l15   = lane & 15;
  const int kHalf = (lane >> 4) * 2;            // 0 or 2
  const int row   = mTile * 16 + l15;
  const int col0  = wave * 64 + l15;            // N base for accumulator 0

  const float* xr = x + (size_t)row * IND;
  v8f acc0 = {}, acc1 = {}, acc2 = {}, acc3 = {};
  #pragma unroll 2
  for (int k = 0; k < IND; k += 4) {
    v2f a;
    a.x = xr[k + kHalf];
    a.y = xr[k + kHalf + 1];
    const float* w0 = W + (size_t)(k + kHalf) * HC + col0;
    const float* w1 = w0 + HC;
    v2f b0, b1, b2, b3;
    b0.x = w0[0];  b0.y = w1[0];
    b1.x = w0[16]; b1.y = w1[16];
    b2.x = w0[32]; b2.y = w1[32];
    b3.x = w0[48]; b3.y = w1[48];
    acc0 = __builtin_amdgcn_wmma_f32_16x16x4_f32(false, a, false, b0, (short)0, acc0, false, false);
    acc1 = __builtin_amdgcn_wmma_f32_16x16x4_f32(false, a, false, b1, (short)0, acc1, false, false);
    acc2 = __builtin_amdgcn_wmma_f32_16x16x4_f32(false, a, false, b2, (short)0, acc2, false, false);
    acc3 = __builtin_amdgcn_wmma_f32_16x16x4_f32(false, a, false, b3, (short)0, acc3, false, false);
  }
  const int rbase = mTile * 16 + (lane >> 4) * 8;
  float* hp = h + (size_t)rbase * HC + col0;
  #pragma unroll
  for (int i = 0; i < 8; ++i) {
    hp[(size_t)i * HC + 0]  = acc0[i];
    hp[(size_t)i * HC + 16] = acc1[i];
    hp[(size_t)i * HC + 32] = acc2[i];
    hp[(size_t)i * HC + 48] = acc3[i];
  }
}

// ------------------------- per-node attention dots -------------------------

__global__ void gat_node_attn_kernel(const float* __restrict__ h,
                                     const float* __restrict__ att_src,
                                     const float* __restrict__ att_dst,
                                     float* __restrict__ a_src,
                                     float* __restrict__ a_dst, int NH) {
  int t = blockIdx.x * blockDim.x + threadIdx.x;   // t = node*8 + head
  if (t >= NH) return;
  int head = t & 7;
  const float* hp = h + (size_t)(t >> 3) * HC + head * CDIM;
  const float* as = att_src + head * CDIM;
  const float* ad = att_dst + head * CDIM;
  float s = 0.0f, d = 0.0f;
  #pragma unroll
  for (int c = 0; c < CDIM; ++c) { float v = hp[c]; s += v * as[c]; d += v * ad[c]; }
  a_src[t] = s;
  a_dst[t] = d;
}

// ------------------------- edge pass A: alpha + segment max ----------------

__global__ void gat_edge_alpha_kernel(const int* __restrict__ srcs,
                                      const int* __restrict__ dsts,
                                      const float* __restrict__ edge_attr,
                                      const float* __restrict__ a_src,
                                      const float* __restrict__ a_dst,
                                      const float* __restrict__ edge_coef,
                                      const float* __restrict__ ea_sum,
                                      float* __restrict__ alpha,
                                      float* __restrict__ amax, int E, int N) {
  int e = blockIdx.x * blockDim.x + threadIdx.x;
  if (e >= E + N) return;
  int s, d; float ea;
  if (e < E) { s = srcs[e]; d = dsts[e]; ea = edge_attr[e]; }
  else       { s = d = e - E; ea = ea_sum[0] / (float)E; }

  const float4* asv = (const float4*)(a_src + (size_t)s * NHEAD);
  const float4* adv = (const float4*)(a_dst + (size_t)d * NHEAD);
  float4 s0 = asv[0], s1 = asv[1];
  float4 d0 = adv[0], d1 = adv[1];
  const float4* cv = (const float4*)edge_coef;
  float4 c0 = cv[0], c1 = cv[1];

  float al[NHEAD];
  al[0] = s0.x + d0.x + ea * c0.x;  al[1] = s0.y + d0.y + ea * c0.y;
  al[2] = s0.z + d0.z + ea * c0.z;  al[3] = s0.w + d0.w + ea * c0.w;
  al[4] = s1.x + d1.x + ea * c1.x;  al[5] = s1.y + d1.y + ea * c1.y;
  al[6] = s1.z + d1.z + ea * c1.z;  al[7] = s1.w + d1.w + ea * c1.w;

  float* ap = alpha + (size_t)e * NHEAD;
  float* mp = amax  + (size_t)d * NHEAD;
  #pragma unroll
  for (int hd = 0; hd < NHEAD; ++hd) {
    float v = al[hd];
    v = (v >= 0.0f) ? v : 0.2f * v;          // leaky_relu, slope 0.2
    ap[hd] = v;
    atomicMaxF(mp + hd, v);
  }
}

// ------------------------- edge pass B: exp + segment sum ------------------

__global__ void gat_edge_exp_kernel(const int* __restrict__ dsts,
                                    float* __restrict__ alpha,
                                    const float* __restrict__ amax,
                                    float* __restrict__ denom, int E, int N) {
  int e = blockIdx.x * blockDim.x + threadIdx.x;
  if (e >= E + N) return;
  int d = (e < E) ? dsts[e] : (e - E);
  float*        ap = alpha + (size_t)e * NHEAD;
  const float*  mp = amax  + (size_t)d * NHEAD;
  float*        dp = denom + (size_t)d * NHEAD;
  #pragma unroll
  for (int hd = 0; hd < NHEAD; ++hd) {
    float ex = __expf(ap[hd] - mp[hd]);
    ap[hd] = ex;                              // alpha buffer now holds exp()
    atomicAdd(dp + hd, ex);
  }
}

// ------------------------- edge pass C: gather/scatter aggregate -----------
// One thread per (edge, head): 8 threads of an edge together stream the full
// contiguous 1KB h[src] row; 32 f32 scatter-atomics per thread (L2 resident).

__global__ void gat_edge_agg_kernel(const int* __restrict__ srcs,
                                    const int* __restrict__ dsts,
                                    const float* __restrict__ ex_buf,
                                    const float* __restrict__ denom,
                                    const float* __restrict__ h,
                                    float* __restrict__ out, int E, int N) {
  int t = blockIdx.x * blockDim.x + threadIdx.x;
  if (t >= (E + N) * NHEAD) return;
  int e    = t >> 3;
  int head = t & 7;
  int s = (e < E) ? srcs[e] : (e - E);
  int d = (e < E) ? dsts[e] : (e - E);
  float ex = ex_buf[(size_t)e * NHEAD + head];
  float dn = denom[(size_t)d * NHEAD + head] + 1e-16f;
  float attn = ex / dn;
  const float4* hp = (const float4*)(h + (size_t)s * HC + head * CDIM);
  float* op = out + (size_t)d * HC + head * CDIM;
  #pragma unroll
  for (int i = 0; i < 8; ++i) {
    float4 v = hp[i];
    atomicAdd(op + i * 4 + 0, v.x * attn);
    atomicAdd(op + i * 4 + 1, v.y * attn);
    atomicAdd(op + i * 4 + 2, v.z * attn);
    atomicAdd(op + i * 4 + 3, v.w * attn);
  }
}

// ------------------------- batchnorm stats ---------------------------------
// block=256: thread t owns column t, rows strided by gridDim -> every row
// read is a fully coalesced 1KB burst; 2 atomics per thread at the end.

__global__ void gat_bn_stats_kernel(const float* __restrict__ out,
                                    float* __restrict__ colsum,
                                    float* __restrict__ colsq, int N) {
  int col = threadIdx.x;
  float s = 0.0f, q = 0.0f;
  for (int r = blockIdx.x; r < N; r += gridDim.x) {
    float v = out[(size_t)r * HC + col];
    s += v; q += v * v;
  }
  atomicAdd(&colsum[col], s);
  atomicAdd(&colsq[col], q);
}

// ------------------------- batchnorm apply + ELU (in place) ----------------
// NOTE: the reference's `bias` add cancels exactly inside BatchNorm
// (mean(out+b) = mean(out)+b, var unchanged), so it is omitted.

__global__ void gat_bn_apply_kernel(float* __restrict__ out,
                                    const float* __restrict__ colsum,
                                    const float* __restrict__ colsq,
                                    const float* __restrict__ gamma,
                                    const float* __restrict__ beta,
                                    int N) {
  int t = blockIdx.x * blockDim.x + threadIdx.x;
  if (t >= N * HC) return;
  int col = t & (HC - 1);
  float invN = 1.0f / (float)N;
  float mean = colsum[col] * invN;
  float var  = colsq[col] * invN - mean * mean;
  float v = (out[t] - mean) * rsqrtf(var + 1e-5f) * gamma[col] + beta[col];
  out[t] = (v > 0.0f) ? v : (__expf(v) - 1.0f);   // ELU
}

// ---------------------------------------------------------------------------

extern "C" void kernel_launch(void* const* d_in, const int* in_sizes, int n_in,
                              void* d_out, int out_size, void* d_ws, size_t ws_size,
                              hipStream_t stream) {
  const float* x         = (const float*)d_in[0];
  const int*   edge_idx  = (const int*)d_in[1];   // [2,E] int32 (JAX default x64-off)
  const float* edge_attr = (const float*)d_in[2];
  const float* W         = (const float*)d_in[3];
  const float* W_edge    = (const float*)d_in[4];
  const float* att_src   = (const float*)d_in[5];
  const float* att_dst   = (const float*)d_in[6];
  const float* att_edge  = (const float*)d_in[7];
  const float* gamma     = (const float*)d_in[9];
  const float* beta      = (const float*)d_in[10];

  const int N  = in_sizes[0] / IND;       // 50000
  const int E  = in_sizes[2];             // 1600000
  const int NH = N * NHEAD;
  const int* srcs = edge_idx;
  const int* dsts = edge_idx + E;

  // ---- workspace carve-up (~110.4 MB) ----
  char* p = (char*)d_ws;
  float* h_buf     = (float*)p;  p += (size_t)N * HC * sizeof(float);
  float* a_src     = (float*)p;  p += (size_t)NH * sizeof(float);
  float* a_dst     = (float*)p;  p += (size_t)NH * sizeof(float);
  float* amax      = (float*)p;  p += (size_t)NH * sizeof(float);
  float* denom     = (float*)p;  p += (size_t)NH * sizeof(float);
  float* alpha     = (float*)p;  p += (size_t)(E + N) * NHEAD * sizeof(float);
  float* colsum    = (float*)p;  p += 256 * sizeof(float);
  float* colsq     = (float*)p;  p += 256 * sizeof(float);
  float* ea_sum    = (float*)p;  p += 256;            // 1 float, padded
  float* edge_coef = (float*)p;  p += 256;            // 8 floats, padded

  float* out = (float*)d_out;                          // f32 accumulation buffer

  // 0) zero the output accumulator (poisoned by harness)
  hipMemsetAsync(out, 0, (size_t)out_size * sizeof(float), stream);

  // 1) init tables (-inf max, zero sums) + edge_coef
  gat_init_kernel<<<1024, 256, 0, stream>>>(amax, denom, colsum, colsq,
                                            ea_sum, edge_coef, W_edge, att_edge, NH);

  // 2) mean(edge_attr) numerator
  gat_ea_reduce_kernel<<<1024, 256, 0, stream>>>(edge_attr, ea_sum, E);

  // 3) h = x @ W  (f32 WMMA 16x16x4, 16x64 tile per wave)
  gat_gemm_wmma_kernel<<<N / 16, 128, 0, stream>>>(x, W, h_buf, N);

  // 4) a_src / a_dst node dots
  gat_node_attn_kernel<<<(NH + 255) / 256, 256, 0, stream>>>(h_buf, att_src, att_dst,
                                                             a_src, a_dst, NH);

  // 5) alpha + segment max (edges + self-loops)
  gat_edge_alpha_kernel<<<(E + N + 255) / 256, 256, 0, stream>>>(
      srcs, dsts, edge_attr, a_src, a_dst, edge_coef, ea_sum, alpha, amax, E, N);

  // 6) exp + segment sum
  gat_edge_exp_kernel<<<(E + N + 255) / 256, 256, 0, stream>>>(
      dsts, alpha, amax, denom, E, N);

  // 7) gather h[src] * attn, scatter-add into out
  gat_edge_agg_kernel<<<((E + N) * NHEAD + 255) / 256, 256, 0, stream>>>(
      srcs, dsts, alpha, denom, h_buf, out, E, N);

  // 8) batchnorm statistics
  gat_bn_stats_kernel<<<1024, 256, 0, stream>>>(out, colsum, colsq, N);

  // 9) normalize + ELU in place
  gat_bn_apply_kernel<<<(N * HC + 255) / 256, 256, 0, stream>>>(
      out, colsum, colsq, gamma, beta, N);
}